// KPInvResidualBlock_66271345377642
// MI455X (gfx1250) — compile-verified
//
#include <hip/hip_runtime.h>
#include <float.h>

typedef __attribute__((ext_vector_type(2))) float v2f;
typedef __attribute__((ext_vector_type(8))) float v8f;

#define NEGS 0.1f
#define EPSB 1e-5f
#define HN   32     // neighbors
#define CMID 32     // mid channels
#define NKP  15     // kernel points
#define NFST 40     // LDS row stride (floats): 2*40 mod 64 = 16 -> halves hit disjoint banks

__device__ __forceinline__ float lrelu(float x) { return x >= 0.f ? x : NEGS * x; }

// ---------------------------------------------------------------------------
// Generic f32 WMMA GEMM: Y(MxN) = X(MxK) @ W(KxN). One wave per 16x16 tile.
// A layout (16x4): lane = M-row (mod 16); VGPR0/1 hold K-pairs, lane halves
// split K {0,1} vs {2,3}. B layout (4x16): lane = N-col; VGPR0/1 = rows,
// halves split rows {0,1} vs {2,3}.
// ---------------------------------------------------------------------------
__global__ void gemm_f32_wmma(const float* __restrict__ X, const float* __restrict__ W,
                              float* __restrict__ Y, int M, int K, int N)
{
    int tilesN = N >> 4;
    int tm = (int)blockIdx.x / tilesN;
    int tn = (int)blockIdx.x % tilesN;
    int lane = threadIdx.x;
    int l16  = lane & 15;
    int off  = (lane < 16) ? 0 : 2;
    int row  = tm * 16 + l16;
    int rowc = row < M ? row : (M - 1);          // clamp for padded tiles
    int col  = tn * 16 + l16;

    v8f acc = {};
    for (int k0 = 0; k0 < K; k0 += 4) {
        v2f a = *(const v2f*)(X + (size_t)rowc * K + k0 + off);
        v2f b;
        b.x = W[(size_t)(k0 + off)     * N + col];
        b.y = W[(size_t)(k0 + off + 1) * N + col];
        acc = __builtin_amdgcn_wmma_f32_16x16x4_f32(false, a, false, b,
                                                    (short)0, acc, false, false);
    }
    int rbase = tm * 16 + ((lane < 16) ? 0 : 8);
#pragma unroll
    for (int v = 0; v < 8; ++v)
        if (rbase + v < M)
            Y[(size_t)(rbase + v) * N + col] = acc[v];
}

// ---------------------------------------------------------------------------
// Column sum / sum-of-squares for batch-norm stats (C divides 256).
// ---------------------------------------------------------------------------
__global__ void colstats(const float* __restrict__ Y, int M, int C,
                         float* __restrict__ sum, float* __restrict__ sumsq)
{
    __shared__ float ls[256], lq[256];
    int t = threadIdx.x;
    int c = t % C;
    int rpb = 256 / C;
    float s = 0.f, q = 0.f;
    for (int row = (int)blockIdx.x * rpb + t / C; row < M; row += (int)gridDim.x * rpb) {
        float v = Y[(size_t)row * C + c];
        s += v; q += v * v;
    }
    ls[t] = s; lq[t] = q;
    __syncthreads();
    if (t < C) {
        for (int j = t + C; j < 256; j += C) { s += ls[j]; q += lq[j]; }
        atomicAdd(&sum[c], s);
        atomicAdd(&sumsq[c], q);
    }
}

__global__ void bn_coeffs(const float* __restrict__ sum, const float* __restrict__ sumsq,
                          const float* __restrict__ g, const float* __restrict__ beta,
                          float* __restrict__ a, float* __restrict__ b, int C, float invM)
{
    int c = threadIdx.x;
    if (c < C) {
        float mean = sum[c] * invM;
        float var  = sumsq[c] * invM - mean * mean;
        float ac   = g[c] * rsqrtf(var + EPSB);
        a[c] = ac;
        b[c] = beta[c] - mean * ac;
    }
}

__global__ void bn_apply_lrelu(const float* __restrict__ Y, const float* __restrict__ a,
                               const float* __restrict__ b, float* __restrict__ Xo,
                               int n, int C)
{
    int i = (int)blockIdx.x * blockDim.x + threadIdx.x;
    if (i < n) { int c = i % C; Xo[i] = lrelu(a[c] * Y[i] + b[c]); }
}

__global__ void final_out(const float* __restrict__ Y2, const float* __restrict__ a,
                          const float* __restrict__ b, const float* __restrict__ sfeat,
                          float* __restrict__ out, int n)
{
    int i = (int)blockIdx.x * blockDim.x + threadIdx.x;
    if (i < n) { int c = i & 127; out[i] = lrelu(a[c] * Y2[i] + b[c] + sfeat[i]); }
}

// ---------------------------------------------------------------------------
// KPInv conv: one wave per point. All LDS state is per-wave, and LDS ops from
// one wave complete in order (DScnt), so no block barriers are needed.
// Per point:
//   - neighbor features gathered straight into LDS via
//     global_load_async_to_lds_b128 (ASYNCcnt path, overlaps with influence
//     computation which only needs positions),
//   fk(16x32) = infl^T(16x32-pad) @ nfeat(32x32) via 16x V_WMMA_F32_16X16X4_F32,
//   dynamic weights w = lrelu(max_h(nfeat) @ wg1 + bg1) @ wg2 + bg2,
//   out[c] = sum_k w[k, c>>4] * fk[k, c]   (group index == WMMA N-tile index).
// ---------------------------------------------------------------------------
__global__ __launch_bounds__(128)
void kpinv_wmma(const float* __restrict__ q_pts, const float* __restrict__ s_pts,
                const float* __restrict__ x, const int* __restrict__ idx,
                const float* __restrict__ kp,
                const float* __restrict__ wg1, const float* __restrict__ bg1,
                const float* __restrict__ wg2, const float* __restrict__ bg2,
                float* __restrict__ conv, int M)
{
    __shared__ __align__(16) float nf[4][HN * NFST]; // gathered neighbor features
    __shared__ float pn[4][HN * 4];      // relative positions + |p|^2
    __shared__ float ctr[4][CMID];       // max-pooled center feature
    __shared__ float hid[4][8];          // hidden layer of weight generator
    __shared__ float wv[4][32];          // generated weights, [k*2+g], zero padded

    int wave = threadIdx.y;
    int lane = threadIdx.x;
    int m = (int)blockIdx.x * 4 + wave;
    if (m >= M) return;                  // wave-uniform: EXEC all-ones below

    float* NF = nf[wave];
    float* PN = pn[wave];

    // --- phase 1: async gather of neighbor features (lane = h) ------------
    int j = idx[(size_t)m * HN + lane];
    {
        unsigned lds_off = (unsigned)(uintptr_t)(NF + lane * NFST);
        unsigned long long ga = (unsigned long long)(uintptr_t)(x + (size_t)j * CMID);
        // same instruction offset applies to both memory and LDS address:
        // both sides advance by 16 bytes per 128-bit chunk of the feature row
        asm volatile(
            "global_load_async_to_lds_b128 %0, %1, off\n\t"
            "global_load_async_to_lds_b128 %0, %1, off offset:16\n\t"
            "global_load_async_to_lds_b128 %0, %1, off offset:32\n\t"
            "global_load_async_to_lds_b128 %0, %1, off offset:48\n\t"
            "global_load_async_to_lds_b128 %0, %1, off offset:64\n\t"
            "global_load_async_to_lds_b128 %0, %1, off offset:80\n\t"
            "global_load_async_to_lds_b128 %0, %1, off offset:96\n\t"
            "global_load_async_to_lds_b128 %0, %1, off offset:112"
            :: "v"(lds_off), "v"(ga) : "memory");
    }
    // relative positions (overlaps with async gather)
    {
        float qx = q_pts[m * 3 + 0], qy = q_pts[m * 3 + 1], qz = q_pts[m * 3 + 2];
        float px = s_pts[j * 3 + 0] - qx;
        float py = s_pts[j * 3 + 1] - qy;
        float pz = s_pts[j * 3 + 2] - qz;
        PN[lane * 4 + 0] = px; PN[lane * 4 + 1] = py;
        PN[lane * 4 + 2] = pz; PN[lane * 4 + 3] = px * px + py * py + pz * pz;
    }

    // --- phase 2: kernel-point influences (lane -> k = lane&15) -----------
    float a_arr[HN];
    int k = lane & 15;
    bool kvalid = (k < NKP);
    float kx = 0.f, ky = 0.f, kz = 0.f, k2 = 0.f;
    if (kvalid) {
        kx = kp[k * 3 + 0]; ky = kp[k * 3 + 1]; kz = kp[k * 3 + 2];
        k2 = kx * kx + ky * ky + kz * kz;
    }
#pragma unroll
    for (int h = 0; h < HN; ++h) {
        float px = PN[h * 4 + 0], py = PN[h * 4 + 1];
        float pz = PN[h * 4 + 2], n2 = PN[h * 4 + 3];
        float d2 = n2 + k2 - 2.f * (px * kx + py * ky + pz * kz);
        d2 = d2 > 0.f ? d2 : 0.f;
        float infl = 1.f - sqrtf(d2);           // SIGMA = 1
        infl = infl > 0.f ? infl : 0.f;
        a_arr[h] = kvalid ? infl : 0.f;          // pad row k=15 with zeros
    }

    // async gather must have landed in LDS before any NF read
    asm volatile("s_wait_asynccnt 0x0" ::: "memory");

    // center max-pool (lane = channel)
    {
        float mx = -FLT_MAX;
        for (int h = 0; h < HN; ++h) {
            float v = NF[h * NFST + lane];
            mx = v > mx ? v : mx;
        }
        ctr[wave][lane] = mx;
    }

    // --- phase 3: dynamic weight generation (in-wave LDS ops are in-order)-
    if (lane < 8) {
        float s = bg1[lane];
        for (int c = 0; c < CMID; ++c) s += ctr[wave][c] * wg1[c * 8 + lane];
        hid[wave][lane] = lrelu(s);
    }
    if (lane < 30) {
        float s = bg2[lane];
#pragma unroll
        for (int jj = 0; jj < 8; ++jj) s += hid[wave][jj] * wg2[jj * 30 + lane];
        wv[wave][lane] = s;
    } else {
        wv[wave][lane] = 0.f;                    // k=15 padding
    }

    // --- phase 4: fk = infl^T @ nfeat via WMMA, then group combine --------
    {
        int off = (lane < 16) ? 0 : 2;
        int n = lane & 15;
        v8f acc0 = {}, acc1 = {};
#pragma unroll
        for (int s4 = 0; s4 < 8; ++s4) {
            v2f a;
            a.x = (lane < 16) ? a_arr[4 * s4 + 0] : a_arr[4 * s4 + 2];
            a.y = (lane < 16) ? a_arr[4 * s4 + 1] : a_arr[4 * s4 + 3];
            int h0 = 4 * s4 + off;
            v2f b0, b1;
            b0.x = NF[h0 * NFST + n];
            b0.y = NF[(h0 + 1) * NFST + n];
            b1.x = NF[h0 * NFST + 16 + n];
            b1.y = NF[(h0 + 1) * NFST + 16 + n];
            acc0 = __builtin_amdgcn_wmma_f32_16x16x4_f32(false, a, false, b0,
                                                         (short)0, acc0, false, false);
            acc1 = __builtin_amdgcn_wmma_f32_16x16x4_f32(false, a, false, b1,
                                                         (short)0, acc1, false, false);
        }
        // D layout: lane holds rows v + (lane<16 ? 0 : 8) of column (tile*16 + n)
        int off8 = (lane < 16) ? 0 : 8;
        float p0 = 0.f, p1 = 0.f;
#pragma unroll
        for (int v = 0; v < 8; ++v) {
            p0 += wv[wave][(v + off8) * 2 + 0] * acc0[v];
            p1 += wv[wave][(v + off8) * 2 + 1] * acc1[v];
        }
        p0 += __shfl_xor(p0, 16, 32);
        p1 += __shfl_xor(p1, 16, 32);
        if (lane < 16) {
            conv[(size_t)m * CMID + n]      = p0;
            conv[(size_t)m * CMID + 16 + n] = p1;
        }
    }
}

// ---------------------------------------------------------------------------
extern "C" void kernel_launch(void* const* d_in, const int* in_sizes, int n_in,
                              void* d_out, int out_size, void* d_ws, size_t ws_size,
                              hipStream_t stream)
{
    const float* q_pts   = (const float*)d_in[0];
    const float* s_pts   = (const float*)d_in[1];
    const float* s_feats = (const float*)d_in[2];
    const int*   idx     = (const int*)  d_in[3];
    const float* kp      = (const float*)d_in[4];
    const float* w_u1    = (const float*)d_in[5];
    const float* g_u1    = (const float*)d_in[6];
    const float* b_u1    = (const float*)d_in[7];
    const float* w_g1    = (const float*)d_in[8];
    const float* b_g1    = (const float*)d_in[9];
    const float* w_g2    = (const float*)d_in[10];
    const float* b_g2    = (const float*)d_in[11];
    const float* g_c     = (const float*)d_in[12];
    const float* b_c     = (const float*)d_in[13];
    const float* w_u2    = (const float*)d_in[14];
    const float* g_u2    = (const float*)d_in[15];
    const float* b_u2    = (const float*)d_in[16];
    float* out = (float*)d_out;

    const int M = in_sizes[0] / 3;
    const int Cin = 128, Cm = CMID;
    const float invM = 1.f / (float)M;
    const int tilesM = (M + 15) / 16;

    float* ws      = (float*)d_ws;
    float* bufA    = ws;                               // M*128 (y1, then y2)
    float* bufX1   = bufA    + (size_t)M * Cin;        // M*32
    float* bufConv = bufX1   + (size_t)M * Cm;         // M*32
    float* bufX2   = bufConv + (size_t)M * Cm;         // M*32
    float* sb      = bufX2   + (size_t)M * Cm;         // 512 floats of stats
    float* s_sum = sb, *s_sq = sb + 128, *s_a = sb + 256, *s_b = sb + 384;

    // ---- stage 1: unary1 = lrelu(bn(s_feats @ w_u1)) ----------------------
    gemm_f32_wmma<<<tilesM * (Cm / 16), 32, 0, stream>>>(s_feats, w_u1, bufA, M, Cin, Cm);
    hipMemsetAsync(sb, 0, 256 * sizeof(float), stream);
    colstats<<<128, 256, 0, stream>>>(bufA, M, Cm, s_sum, s_sq);
    bn_coeffs<<<1, 128, 0, stream>>>(s_sum, s_sq, g_u1, b_u1, s_a, s_b, Cm, invM);
    bn_apply_lrelu<<<(M * Cm + 255) / 256, 256, 0, stream>>>(bufA, s_a, s_b, bufX1, M * Cm, Cm);

    // ---- stage 2: KPInv conv, then lrelu(bn(.)) ---------------------------
    kpinv_wmma<<<(M + 3) / 4, dim3(32, 4), 0, stream>>>(q_pts, s_pts, bufX1, idx, kp,
                                                        w_g1, b_g1, w_g2, b_g2, bufConv, M);
    hipMemsetAsync(sb, 0, 256 * sizeof(float), stream);
    colstats<<<128, 256, 0, stream>>>(bufConv, M, Cm, s_sum, s_sq);
    bn_coeffs<<<1, 128, 0, stream>>>(s_sum, s_sq, g_c, b_c, s_a, s_b, Cm, invM);
    bn_apply_lrelu<<<(M * Cm + 255) / 256, 256, 0, stream>>>(bufConv, s_a, s_b, bufX2, M * Cm, Cm);

    // ---- stage 3: unary2 = bn(x2 @ w_u2); out = lrelu(. + s_feats) -------
    gemm_f32_wmma<<<tilesM * (Cin / 16), 32, 0, stream>>>(bufX2, w_u2, bufA, M, Cm, Cin);
    hipMemsetAsync(sb, 0, 256 * sizeof(float), stream);
    colstats<<<128, 256, 0, stream>>>(bufA, M, Cin, s_sum, s_sq);
    bn_coeffs<<<1, 128, 0, stream>>>(s_sum, s_sq, g_u2, b_u2, s_a, s_b, Cin, invM);
    final_out<<<(M * Cin + 255) / 256, 256, 0, stream>>>(bufA, s_a, s_b, s_feats, out, M * Cin);
}